// SurvivalCLMBRTask_90443421319262
// MI455X (gfx1250) — compile-verified
//
#include <hip/hip_runtime.h>
#include <hip/hip_bf16.h>
#include <math.h>

typedef __attribute__((ext_vector_type(2))) float v2f;
typedef __attribute__((ext_vector_type(8))) float v8f;

#define MROWS 8192   // B*T
#define NCODES 8192
#define KDIM 512     // dim
#define BSZ 1024
#define TBINS 8
#define FDIM 768
#define NW 4088      // T*(dim-1)
#define DM1 511
#define NNZ_CT 65536
#define NEV_CT 16384

__device__ __forceinline__ v8f wmma4(v2f a, v2f b, v8f c) {
  // D = A(16x4,f32) x B(4x16,f32) + C(16x16,f32) -- v_wmma_f32_16x16x4_f32
  return __builtin_amdgcn_wmma_f32_16x16x4_f32(false, a, false, b, (short)0, c,
                                               false, false);
}

// ---------------------------------------------------------------------------
// GEMM2: logits = exp2(full_a @ tcw^T), fused row-sum of exp2 into rowSum[].
// 128x128 tile / workgroup, 8 waves, each wave = 32(M) x 64(N) = 2x4 frags.
// ---------------------------------------------------------------------------
__global__ __launch_bounds__(256) void gemm2_logits_kernel(
    const float* __restrict__ A, const float* __restrict__ Bm,
    float* __restrict__ logits, float* __restrict__ rowSum) {
  __shared__ float As[16][128];
  __shared__ float Bs[16][128];
  const int tid = threadIdx.x;
  const int lane = tid & 31;
  const int wave = tid >> 5;
  const int half = lane >> 4;   // K-pair select for A/B frags; row+8 for C/D
  const int idx = lane & 15;    // M row (A) / N col (B, C/D)
  const int m0 = blockIdx.y << 7;
  const int n0 = blockIdx.x << 7;
  const int mbase = (wave >> 1) << 5;  // wave M offset in tile
  const int nbase = (wave & 1) << 6;   // wave N offset in tile

  v8f acc[2][4];
#pragma unroll
  for (int i = 0; i < 2; ++i)
#pragma unroll
    for (int j = 0; j < 4; ++j)
      acc[i][j] = (v8f){0.f, 0.f, 0.f, 0.f, 0.f, 0.f, 0.f, 0.f};

  for (int kc = 0; kc < KDIM; kc += 16) {
    // Stage 128x16 of A and B into LDS, k-major for fragment loads.
#pragma unroll
    for (int s = tid; s < 512; s += 256) {
      const int mr = s >> 2;
      const int kg = (s & 3) << 2;
      const float4 av =
          *reinterpret_cast<const float4*>(A + (size_t)(m0 + mr) * KDIM + kc + kg);
      As[kg + 0][mr] = av.x; As[kg + 1][mr] = av.y;
      As[kg + 2][mr] = av.z; As[kg + 3][mr] = av.w;
      const float4 bv =
          *reinterpret_cast<const float4*>(Bm + (size_t)(n0 + mr) * KDIM + kc + kg);
      Bs[kg + 0][mr] = bv.x; Bs[kg + 1][mr] = bv.y;
      Bs[kg + 2][mr] = bv.z; Bs[kg + 3][mr] = bv.w;
    }
    __syncthreads();
#pragma unroll
    for (int ks = 0; ks < 16; ks += 4) {
      const int kb = ks + (half << 1);
      v2f a0 = {As[kb][mbase + idx], As[kb + 1][mbase + idx]};
      v2f a1 = {As[kb][mbase + 16 + idx], As[kb + 1][mbase + 16 + idx]};
#pragma unroll
      for (int nf = 0; nf < 4; ++nf) {
        v2f b = {Bs[kb][nbase + (nf << 4) + idx],
                 Bs[kb + 1][nbase + (nf << 4) + idx]};
        acc[0][nf] = wmma4(a0, b, acc[0][nf]);
        acc[1][nf] = wmma4(a1, b, acc[1][nf]);
      }
    }
    __syncthreads();
  }

  // Epilogue: exp2, store logits, reduce row sums (lane=col, vgpr r=row).
#pragma unroll
  for (int mf = 0; mf < 2; ++mf) {
    float rs[8];
#pragma unroll
    for (int r = 0; r < 8; ++r) rs[r] = 0.0f;
#pragma unroll
    for (int nf = 0; nf < 4; ++nf) {
      const int col = n0 + nbase + (nf << 4) + idx;
#pragma unroll
      for (int r = 0; r < 8; ++r) {
        const int row = m0 + mbase + (mf << 4) + (half << 3) + r;
        const float e = exp2f(acc[mf][nf][r]);
        logits[(size_t)row * NCODES + col] = e;
        rs[r] += e;
      }
    }
#pragma unroll
    for (int r = 0; r < 8; ++r) {
      float s = rs[r];
#pragma unroll
      for (int off = 8; off >= 1; off >>= 1) s += __shfl_xor(s, off, 16);
      if (idx == 0) {
        const int row = m0 + mbase + (mf << 4) + (half << 3) + r;
        atomicAdd(rowSum + row, s);
      }
    }
  }
}

// ---------------------------------------------------------------------------
// GEMM1: (features @ final_W + final_b) reshaped into full_a[M, 512] layout.
// N padded 4088 -> 4096, guarded B loads / stores.
// ---------------------------------------------------------------------------
__global__ __launch_bounds__(256) void gemm1_kernel(
    const float* __restrict__ feat, const float* __restrict__ W,
    const float* __restrict__ bias, float* __restrict__ fullA) {
  __shared__ float As[16][128];
  __shared__ float Bs[16][128];
  const int tid = threadIdx.x;
  const int lane = tid & 31;
  const int wave = tid >> 5;
  const int half = lane >> 4;
  const int idx = lane & 15;
  const int m0 = blockIdx.y << 7;
  const int n0 = blockIdx.x << 7;
  const int mbase = (wave >> 1) << 5;
  const int nbase = (wave & 1) << 6;

  v8f acc[2][4];
#pragma unroll
  for (int i = 0; i < 2; ++i)
#pragma unroll
    for (int j = 0; j < 4; ++j)
      acc[i][j] = (v8f){0.f, 0.f, 0.f, 0.f, 0.f, 0.f, 0.f, 0.f};

  for (int kc = 0; kc < FDIM; kc += 16) {
#pragma unroll
    for (int s = tid; s < 512; s += 256) {
      const int mr = s >> 2;
      const int kg = (s & 3) << 2;
      const float4 av = *reinterpret_cast<const float4*>(
          feat + (size_t)(m0 + mr) * FDIM + kc + kg);
      As[kg + 0][mr] = av.x; As[kg + 1][mr] = av.y;
      As[kg + 2][mr] = av.z; As[kg + 3][mr] = av.w;
    }
#pragma unroll
    for (int s = tid; s < 2048; s += 256) {
      const int k = s >> 7;
      const int n = s & 127;
      const int gn = n0 + n;
      Bs[k][n] = (gn < NW) ? W[(size_t)(kc + k) * NW + gn] : 0.0f;
    }
    __syncthreads();
#pragma unroll
    for (int ks = 0; ks < 16; ks += 4) {
      const int kb = ks + (half << 1);
      v2f a0 = {As[kb][mbase + idx], As[kb + 1][mbase + idx]};
      v2f a1 = {As[kb][mbase + 16 + idx], As[kb + 1][mbase + 16 + idx]};
#pragma unroll
      for (int nf = 0; nf < 4; ++nf) {
        v2f b = {Bs[kb][nbase + (nf << 4) + idx],
                 Bs[kb + 1][nbase + (nf << 4) + idx]};
        acc[0][nf] = wmma4(a0, b, acc[0][nf]);
        acc[1][nf] = wmma4(a1, b, acc[1][nf]);
      }
    }
    __syncthreads();
  }

#pragma unroll
  for (int mf = 0; mf < 2; ++mf) {
#pragma unroll
    for (int nf = 0; nf < 4; ++nf) {
      const int col = n0 + nbase + (nf << 4) + idx;
      if (col < NW) {
        const float bcol = bias[col];
        const int t = col / DM1;
        const int d = col - t * DM1;
#pragma unroll
        for (int r = 0; r < 8; ++r) {
          const int m = m0 + mbase + (mf << 4) + (half << 3) + r;
          fullA[(size_t)(m * TBINS + t) * KDIM + d] = acc[mf][nf][r] + bcol;
        }
      }
    }
  }
}

// ---------------------------------------------------------------------------
// Small helper kernels
// ---------------------------------------------------------------------------
__global__ void prep_tcw_kernel(const float* __restrict__ cw,
                                const float* __restrict__ cwb,
                                float* __restrict__ tcw) {
  const int i = blockIdx.x * blockDim.x + threadIdx.x;
  if (i < NCODES * KDIM) {
    const int n = i >> 9;
    const int k = i & (KDIM - 1);
    tcw[i] = (k < DM1) ? cw[(size_t)n * DM1 + k] : cwb[n];
  }
}

__global__ void init_kernel(float* __restrict__ fullA, float* __restrict__ rowSum,
                            float* __restrict__ accum) {
  const int i = blockIdx.x * blockDim.x + threadIdx.x;
  if (i < MROWS) {
    fullA[(size_t)i * KDIM + DM1] = 1.0f;  // ones column of full_a
    rowSum[i] = 0.0f;
  }
  if (i < 8) accum[i] = 0.0f;
}

__global__ void reduce_base_kernel(const float* __restrict__ rowSum,
                                   const float* __restrict__ defaults,
                                   float* __restrict__ accum) {
  const int i = blockIdx.x * blockDim.x + threadIdx.x;
  float s = (i < MROWS) ? exp2f(defaults[i]) * rowSum[i] : 0.0f;
#pragma unroll
  for (int off = 16; off >= 1; off >>= 1) s += __shfl_xor(s, off, 32);
  if ((threadIdx.x & 31) == 0) atomicAdd(&accum[0], s);
}

__device__ __forceinline__ float wave_dot512(const float* __restrict__ a,
                                             const float* __restrict__ b,
                                             int lane) {
  float s = 0.0f;
#pragma unroll
  for (int i = 0; i < 4; ++i) {
    const float4 x = reinterpret_cast<const float4*>(const_cast<float*>(a))[lane + (i << 5)];
    const float4 y = reinterpret_cast<const float4*>(const_cast<float*>(b))[lane + (i << 5)];
    s = fmaf(x.x, y.x, s); s = fmaf(x.y, y.y, s);
    s = fmaf(x.z, y.z, s); s = fmaf(x.w, y.w, s);
  }
#pragma unroll
  for (int off = 16; off >= 1; off >>= 1) s += __shfl_xor(s, off, 32);
  return s;
}

__global__ __launch_bounds__(256) void sparse_corr_kernel(
    const float* __restrict__ fullA, const float* __restrict__ tcw,
    const int* __restrict__ rows, const int* __restrict__ cols,
    const float* __restrict__ defaults, const float* __restrict__ values,
    float* __restrict__ accum) {
  const int gw = (int)((blockIdx.x * blockDim.x + threadIdx.x) >> 5);
  const int lane = threadIdx.x & 31;
  if (gw >= NNZ_CT) return;
  const int r = rows[gw];
  const int c = cols[gw];
  const float d = wave_dot512(fullA + (size_t)r * KDIM, tcw + (size_t)c * KDIM, lane);
  if (lane == 0) {
    const float contrib = exp2f(d + values[gw]) - exp2f(d + defaults[r]);
    atomicAdd(&accum[1], contrib);
  }
}

__global__ __launch_bounds__(256) void event_kernel(
    const float* __restrict__ fullA, const float* __restrict__ tcw,
    const int* __restrict__ ev, float* __restrict__ accum) {
  const int gw = (int)((blockIdx.x * blockDim.x + threadIdx.x) >> 5);
  const int lane = threadIdx.x & 31;
  if (gw >= NEV_CT) return;
  const int r = ev[2 * gw];
  const int c = ev[2 * gw + 1];
  const float d = wave_dot512(fullA + (size_t)r * KDIM, tcw + (size_t)c * KDIM, lane);
  if (lane == 0) atomicAdd(&accum[2], d);
}

__global__ void mask_count_kernel(const unsigned char* __restrict__ mask,
                                  float* __restrict__ accum) {
  const int i = blockIdx.x * blockDim.x + threadIdx.x;
  float s = (i < BSZ && mask[i]) ? 1.0f : 0.0f;
#pragma unroll
  for (int off = 16; off >= 1; off >>= 1) s += __shfl_xor(s, off, 32);
  if ((threadIdx.x & 31) == 0) atomicAdd(&accum[3], s);
}

__global__ void finalize_kernel(const float* __restrict__ accum,
                                float* __restrict__ out) {
  if (threadIdx.x == 0 && blockIdx.x == 0) {
    const float base = accum[0];
    const float corr = accum[1];
    const float ev = accum[2];
    const float nm = accum[3];
    const float MN = (float)MROWS * (float)NCODES;
    const float exp_mean = (base + corr) / MN;
    const float survival = exp_mean * ((float)MROWS / nm);
    const float event_loss = -0.69314718055994530942f * ev / (nm * (float)NCODES);
    out[0] = event_loss + survival;
  }
}

// ---------------------------------------------------------------------------
extern "C" void kernel_launch(void* const* d_in, const int* in_sizes, int n_in,
                              void* d_out, int out_size, void* d_ws, size_t ws_size,
                              hipStream_t stream) {
  const float* features = (const float*)d_in[0];
  const unsigned char* mask = (const unsigned char*)d_in[1];  // jax bool -> 1B
  const float* code_weight = (const float*)d_in[2];
  const float* code_weight_bias = (const float*)d_in[3];
  const float* final_W = (const float*)d_in[4];
  const float* final_b = (const float*)d_in[5];
  const int* sparse_rows = (const int*)d_in[6];
  const int* sparse_cols = (const int*)d_in[7];
  const float* sparse_defaults = (const float*)d_in[8];
  const float* sparse_values = (const float*)d_in[9];
  const int* event_indices = (const int*)d_in[10];

  float* out = (float*)d_out;
  float* logits = out + 1;  // out[0] = scalar loss, then logits [8192, 8192]

  float* ws = (float*)d_ws;
  float* fullA = ws;                                // MROWS * KDIM
  float* tcw = fullA + (size_t)MROWS * KDIM;        // NCODES * KDIM
  float* rowSum = tcw + (size_t)NCODES * KDIM;      // MROWS
  float* accum = rowSum + MROWS;                    // 8 scalars

  prep_tcw_kernel<<<(NCODES * KDIM + 255) / 256, 256, 0, stream>>>(
      code_weight, code_weight_bias, tcw);
  init_kernel<<<(MROWS + 255) / 256, 256, 0, stream>>>(fullA, rowSum, accum);
  gemm1_kernel<<<dim3(32, 8), 256, 0, stream>>>(features, final_W, final_b, fullA);
  gemm2_logits_kernel<<<dim3(64, 64), 256, 0, stream>>>(fullA, tcw, logits, rowSum);
  reduce_base_kernel<<<(MROWS + 255) / 256, 256, 0, stream>>>(rowSum,
                                                              sparse_defaults, accum);
  sparse_corr_kernel<<<(NNZ_CT * 32 + 255) / 256, 256, 0, stream>>>(
      fullA, tcw, sparse_rows, sparse_cols, sparse_defaults, sparse_values, accum);
  event_kernel<<<(NEV_CT * 32 + 255) / 256, 256, 0, stream>>>(fullA, tcw,
                                                              event_indices, accum);
  mask_count_kernel<<<(BSZ + 255) / 256, 256, 0, stream>>>(mask, accum);
  finalize_kernel<<<1, 1, 0, stream>>>(accum, out);
}